// EQL_63496796504639
// MI455X (gfx1250) — compile-verified
//
#include <hip/hip_runtime.h>
#include <hip/hip_bf16.h>
#include <math.h>

typedef __attribute__((ext_vector_type(2))) float v2f;
typedef __attribute__((ext_vector_type(8))) float v8f;

#define NUM_INPUTS 64
#define NUM_OUT    15
#define SAMPLES    16
#define BATCH      4096
#define WSHAPE     3120
#define BSHAPE     41
#define WLAST_     3026
#define BLAST_     40
#define INFINAL    94
#define TEMP_      0.03f
#define EPSF       1e-20f
#define OUT_ELEMS  (SAMPLES*BATCH*NUM_OUT)

#define BT      64            // batch rows per workgroup
#define NWAVE   (BT/16)       // 4 waves (wave32)
#define THREADS (NWAVE*32)    // 128 threads
#define HSTRIDE 100           // 100 mod 64 = 36 -> 16 distinct banks; rows 16B-aligned
#define SSTRIDE 17
#define NCHUNK  116           // sum of ceil(d/4) over the 6 layers
#define SO_STRIDE 7520        // floats per (s,o) ws region: 116*64 frag + 94 fin (pad 7520)

__device__ constexpr int L_D[6]   = {64,67,70,78,86,90};
__device__ constexpr int L_E[6]   = {6,6,8,8,6,6};
__device__ constexpr int L_NOP[6] = {3,3,8,8,4,4};
__device__ constexpr int L_WLO[6] = {0,384,786,1346,1970,2486};
__device__ constexpr int L_BLO[6] = {0,6,12,20,28,34};
__device__ constexpr int L_PCH[6] = {0,16,33,51,71,93};   // chunk base per layer
__device__ constexpr int OPTBL[6][8] = {
  {0,0,0,0,0,0,0,0},
  {1,1,1,0,0,0,0,0},
  {2,2,3,3,4,4,5,5},
  {2,2,3,3,4,4,5,5},
  {0,1,2,3,0,0,0,0},
  {0,1,2,3,0,0,0,0}};
__device__ constexpr int OPIN[8] = {2,2,1,1,1,1,1,3};

// chunk index -> layer index (116 chunks total)
__device__ __forceinline__ int chunk_layer(int c) {
  return (c < 16) ? 0 : (c < 33) ? 1 : (c < 51) ? 2 : (c < 71) ? 3 : (c < 93) ? 4 : 5;
}

// hard-concrete gate: w = constw_base * ((hard - soft) + soft)  (forward value)
__device__ __forceinline__ float gate_weight(float sc, float cwb, float uu0, float uu1) {
  float l0    = __logf(uu0 + EPSF);
  float l1    = __logf(uu1 + EPSF);
  float noise = -__logf(l0 / l1 + EPSF);
  float cs    = 1.0f / (1.0f + __expf(-sc));
  float logit = __logf(cs + EPSF) - __logf(1.0f - cs + EPSF);
  float soft  = 1.0f / (1.0f + __expf(-(logit + noise) * TEMP_));
  float hard  = soft > 0.5f ? 1.0f : 0.0f;
  return cwb * ((hard - soft) + soft);
}

__device__ __forceinline__ float apply_op(int idx, const float* h, float& regu) {
  switch (idx) {
    case 0: { // mul
      float a = h[0], b = h[1];
      regu += fmaxf(-100.0f - a, 0.0f) + fmaxf(a - 100.0f, 0.0f)
            + fmaxf(-100.0f - b, 0.0f) + fmaxf(b - 100.0f, 0.0f);
      return fminf(fmaxf(a, -100.0f), 100.0f) * fminf(fmaxf(b, -100.0f), 100.0f);
    }
    case 1: { // div
      float a = h[0], b = h[1];
      regu += fmaxf(0.01f - b, 0.0f);
      return (b < 0.01f) ? 0.0f : __fdividef(a, b);
    }
    case 2: { // log
      float a = h[0];
      regu += fmaxf(0.001f - a, 0.0f);
      return __logf(fmaxf(a, 0.001f));
    }
    case 3: { // exp
      float a = h[0];
      regu += fmaxf(-10.0f - a, 0.0f) + fmaxf(a - 4.0f, 0.0f);
      return __expf(fminf(fmaxf(a, -10.0f), 4.0f));
    }
    case 4: return __sinf(h[0]);
    case 5: return __cosf(h[0]);
    case 6: return h[0];
    default: { // ifelse
      float g = 1.0f / (1.0f + __expf(-h[0]));
      return g * h[1] + (1.0f - g) * h[2];
    }
  }
}

// Gate kernel: one block per (s,o). Writes WMMA-B-fragment-swizzled gated
// weights (zero padding baked in) + final-layer weights into ws.
__global__ __launch_bounds__(256)
void eql_gate_kernel(const float* __restrict__ scores,
                     const float* __restrict__ cwb,
                     const float* __restrict__ u0,
                     const float* __restrict__ u1,
                     float* __restrict__ ws) {
  const int so  = blockIdx.x;           // 0..239
  const int o   = so % NUM_OUT;
  const int tid = threadIdx.x;
  const float* sc = scores + o * WSHAPE;
  const float* cw = cwb    + o * WSHAPE;
  const float* a0 = u0 + (size_t)so * WSHAPE;
  const float* a1 = u1 + (size_t)so * WSHAPE;
  float* dst = ws + (size_t)so * SO_STRIDE;

  auto gw = [&](int i) { return gate_weight(sc[i], cw[i], a0[i], a1[i]); };

  for (int idx = tid; idx < NCHUNK * 32; idx += 256) {
    const int cidx = idx >> 5, L = idx & 31;
    const int li = chunk_layer(cidx);
    const int d = L_D[li], e = L_E[li], wlo = L_WLO[li];
    const int kc = cidx - L_PCH[li];
    const int n  = L & 15;
    const int k0 = 4 * kc + ((L >> 4) << 1);
    v2f v;
    v.x = (k0     < d && n < e) ? gw(wlo + k0       * e + n) : 0.0f;
    v.y = (k0 + 1 < d && n < e) ? gw(wlo + (k0 + 1) * e + n) : 0.0f;
    ((v2f*)dst)[idx] = v;
  }
  for (int i = tid; i < INFINAL; i += 256) dst[NCHUNK * 64 + i] = gw(WLAST_ + i);
}

__global__ void eql_zero_kernel(float* __restrict__ out) {
  if (blockIdx.x == 0 && threadIdx.x == 0) out[OUT_ELEMS] = 0.0f;
}

__global__ __launch_bounds__(THREADS)
void eql_main_kernel(const float* __restrict__ obs,
                     const float* __restrict__ scores,
                     const float* __restrict__ cwb,
                     const float* __restrict__ cb,
                     const float* __restrict__ u0,
                     const float* __restrict__ u1,
                     const float* __restrict__ wpre,   // nullptr -> compute inline
                     float* __restrict__ out) {
  __shared__ float hid[BT * HSTRIDE];                    // 25600 B
  __shared__ __attribute__((aligned(16))) v2f   wfrag[NCHUNK * 32];  // 29696 B
  __shared__ __attribute__((aligned(16))) float wfin[96];
  __shared__ float b_lds[BSHAPE];
  __shared__ float stage[NWAVE * 16 * SSTRIDE];

  const int tid   = threadIdx.x;
  const int btile = blockIdx.x;
  const int o     = blockIdx.y;
  const int s     = blockIdx.z;
  const int so    = s * NUM_OUT + o;

  // ---- weights into LDS ----
  if (wpre) {
    // primary path: plain vectorized copy of pre-swizzled fragments
    const float4* src = (const float4*)(wpre + (size_t)so * SO_STRIDE);
    float4* dst = (float4*)wfrag;
    for (int i = tid; i < (NCHUNK * 64) / 4; i += THREADS) dst[i] = src[i];
    const float* fsrc = wpre + (size_t)so * SO_STRIDE + NCHUNK * 64;
    for (int i = tid; i < 96; i += THREADS) wfin[i] = (i < INFINAL) ? fsrc[i] : 0.0f;
  } else {
    // fallback: compute gates inline (generic, non-unrolled loop)
    const float* sc = scores + o * WSHAPE;
    const float* cw = cwb    + o * WSHAPE;
    const float* a0 = u0 + (size_t)so * WSHAPE;
    const float* a1 = u1 + (size_t)so * WSHAPE;
    auto gw = [&](int i) { return gate_weight(sc[i], cw[i], a0[i], a1[i]); };
    for (int idx = tid; idx < NCHUNK * 32; idx += THREADS) {
      const int cidx = idx >> 5, L = idx & 31;
      const int li = chunk_layer(cidx);
      const int d = L_D[li], e = L_E[li], wlo = L_WLO[li];
      const int kc = cidx - L_PCH[li];
      const int n  = L & 15;
      const int k0 = 4 * kc + ((L >> 4) << 1);
      v2f v;
      v.x = (k0     < d && n < e) ? gw(wlo + k0       * e + n) : 0.0f;
      v.y = (k0 + 1 < d && n < e) ? gw(wlo + (k0 + 1) * e + n) : 0.0f;
      wfrag[idx] = v;
    }
    for (int i = tid; i < 96; i += THREADS) wfin[i] = (i < INFINAL) ? gw(WLAST_ + i) : 0.0f;
  }
  if (tid < BSHAPE) b_lds[tid] = cb[o * BSHAPE + tid];

  // ---- obs tile -> hid[:,0:64] (b128 both sides); zero cols 64..99 ----
  {
    const float4* ob4 = (const float4*)(obs + (size_t)(btile * BT) * NUM_INPUTS);
    for (int i = tid; i < BT * (NUM_INPUTS / 4); i += THREADS) {
      const int r = i >> 4, c4 = i & 15;
      *(float4*)&hid[r * HSTRIDE + c4 * 4] = ob4[i];
    }
    const int PADC = HSTRIDE - NUM_INPUTS; // 36
    for (int i = tid; i < BT * PADC; i += THREADS) {
      const int r = i / PADC, c = NUM_INPUTS + (i % PADC);
      hid[r * HSTRIDE + c] = 0.0f;
    }
  }
  __syncthreads();

  const int wave = tid >> 5;
  const int lane = tid & 31;
  const int m    = lane & 15;   // A: row M; B/D: column N
  const int hi   = lane >> 4;   // half-wave selector
  float regu = 0.0f;

  const float* arow = hid + (wave * 16 + m) * HSTRIDE;   // 8B-aligned at even k0
  const v2f*   bcol = wfrag + lane;

  #pragma unroll
  for (int li = 0; li < 6; ++li) {
    const int d   = L_D[li];
    const int e   = L_E[li];
    const int blo = L_BLO[li];
    const int kch = (d + 3) >> 2;
    const int pb  = L_PCH[li];

    // two accumulator chains -> 2x ILP on the matrix pipe
    v8f acc0 = {0.f,0.f,0.f,0.f,0.f,0.f,0.f,0.f};
    v8f acc1 = {0.f,0.f,0.f,0.f,0.f,0.f,0.f,0.f};
    int kc = 0;
    for (; kc + 1 < kch; kc += 2) {
      v2f a0v = *(const v2f*)&arow[4 * kc + 2 * hi];
      v2f b0v = bcol[(pb + kc) * 32];
      acc0 = __builtin_amdgcn_wmma_f32_16x16x4_f32(false, a0v, false, b0v,
                                                   (short)0, acc0, false, false);
      v2f a1v = *(const v2f*)&arow[4 * (kc + 1) + 2 * hi];
      v2f b1v = bcol[(pb + kc + 1) * 32];
      acc1 = __builtin_amdgcn_wmma_f32_16x16x4_f32(false, a1v, false, b1v,
                                                   (short)0, acc1, false, false);
    }
    if (kch & 1) {
      v2f a0v = *(const v2f*)&arow[4 * kc + 2 * hi];
      v2f b0v = bcol[(pb + kc) * 32];
      acc0 = __builtin_amdgcn_wmma_f32_16x16x4_f32(false, a0v, false, b0v,
                                                   (short)0, acc0, false, false);
    }

    // D + bias -> stage (C/D layout: VGPR r = row r + 8*hi, col = lane&15)
    if (m < e) {
      const float bias = b_lds[blo + m];
      float* st = stage + wave * 16 * SSTRIDE + hi * 8 * SSTRIDE + m;
      #pragma unroll
      for (int r = 0; r < 8; ++r) st[r * SSTRIDE] = acc0[r] + acc1[r] + bias;
    }
    asm volatile("s_wait_dscnt 0" ::: "memory");

    // elementwise ops: lanes 0..15 each own one batch row; append to hidden
    if (lane < 16) {
      const float* st = stage + wave * 16 * SSTRIDE + lane * SSTRIDE;
      float* ho = hid + (wave * 16 + lane) * HSTRIDE;
      int off = 0;
      #pragma unroll
      for (int j = 0; j < 8; ++j) {
        if (j < L_NOP[li]) {
          const int idx = OPTBL[li][j];
          ho[d + j] = apply_op(idx, st + off, regu);
          off += OPIN[idx];
        }
      }
    }
    asm volatile("s_wait_dscnt 0" ::: "memory");
  }

  // ---- final 94 -> 1 projection: half-waves at col 0 / col 48, b128 loads
  //      (hid cols 94..99 and wfin[94..95] are zero) ----
  {
    const float* hrow = hid + (wave * 16 + m) * HSTRIDE;
    const int c0 = hi * 48;
    const float4* h4 = (const float4*)(hrow + c0);
    const float4* w4 = (const float4*)(wfin + c0);
    float accf = 0.0f;
    #pragma unroll
    for (int c4 = 0; c4 < 12; ++c4) {
      float4 hv = h4[c4], wv = w4[c4];
      accf = fmaf(hv.x, wv.x, accf);
      accf = fmaf(hv.y, wv.y, accf);
      accf = fmaf(hv.z, wv.z, accf);
      accf = fmaf(hv.w, wv.w, accf);
    }
    accf += __shfl_xor(accf, 16, 32);
    if (lane < 16) {
      const int b = btile * BT + wave * 16 + lane;
      out[((size_t)s * BATCH + b) * NUM_OUT + o] = accf + b_lds[BLAST_];
    }
  }

  // ---- regu: wave reduce, one atomic per wave ----
  #pragma unroll
  for (int off = 16; off > 0; off >>= 1)
    regu += __shfl_xor(regu, off, 32);
  if (lane == 0)
    atomicAdd(out + OUT_ELEMS, regu * (1.0f / (float)OUT_ELEMS));
}

extern "C" void kernel_launch(void* const* d_in, const int* in_sizes, int n_in,
                              void* d_out, int out_size, void* d_ws, size_t ws_size,
                              hipStream_t stream) {
  const float* obs    = (const float*)d_in[0];
  const float* scores = (const float*)d_in[1];
  const float* cwb    = (const float*)d_in[2];
  const float* cb     = (const float*)d_in[3];
  const float* u0     = (const float*)d_in[4];
  const float* u1     = (const float*)d_in[5];
  float* out = (float*)d_out;

  const float* wpre = nullptr;
  const size_t need = (size_t)SAMPLES * NUM_OUT * SO_STRIDE * sizeof(float);
  if (d_ws != nullptr && ws_size >= need) {
    eql_gate_kernel<<<SAMPLES * NUM_OUT, 256, 0, stream>>>(scores, cwb, u0, u1,
                                                           (float*)d_ws);
    wpre = (const float*)d_ws;
  }

  eql_zero_kernel<<<1, 64, 0, stream>>>(out);

  dim3 grid(BATCH / BT, NUM_OUT, SAMPLES);
  eql_main_kernel<<<grid, dim3(THREADS), 0, stream>>>(obs, scores, cwb, cb,
                                                      u0, u1, wpre, out);
}